// DropEdgeGraphSAGE_50680614093676
// MI455X (gfx1250) — compile-verified
//
#include <hip/hip_runtime.h>

#define HID 128
#define D_IN 16
#define D_OUT 4
#define LN_EPS 1e-5f

typedef float v2f __attribute__((ext_vector_type(2)));
typedef float v8f __attribute__((ext_vector_type(8)));

// ---------------------------------------------------------------- utilities
__global__ void zero_kernel(float* __restrict__ p, int n) {
  int i = blockIdx.x * blockDim.x + threadIdx.x;
  int stride = gridDim.x * blockDim.x;
  for (; i < n; i += stride) p[i] = 0.0f;
}

__global__ void deg_kernel(const int* __restrict__ dst, float* __restrict__ deg, int E) {
  int e = blockIdx.x * blockDim.x + threadIdx.x;
  if (e < E) unsafeAtomicAdd(&deg[dst[e]], 1.0f);
}

__global__ void invdeg_kernel(float* __restrict__ deg, int N) {
  int i = blockIdx.x * blockDim.x + threadIdx.x;
  if (i < N) {
    float d = deg[i];
    deg[i] = 1.0f / fmaxf(d, 1.0f);
  }
}

// ------------------------------------------------- input projection (K=16)
__global__ void in_proj_kernel(const float* __restrict__ x,
                               const float* __restrict__ W_in,
                               const float* __restrict__ b_in,
                               float* __restrict__ h, int N) {
  int idx = blockIdx.x * blockDim.x + threadIdx.x;
  if (idx >= N * HID) return;
  int row = idx >> 7;
  int col = idx & (HID - 1);
  const float4* xr = (const float4*)(x + (size_t)row * D_IN);
  float acc = b_in[col];
#pragma unroll
  for (int kq = 0; kq < D_IN / 4; ++kq) {
    float4 xv = xr[kq];
    acc += xv.x * W_in[(kq * 4 + 0) * HID + col];
    acc += xv.y * W_in[(kq * 4 + 1) * HID + col];
    acc += xv.z * W_in[(kq * 4 + 2) * HID + col];
    acc += xv.w * W_in[(kq * 4 + 3) * HID + col];
  }
  h[idx] = acc;
}

// ------------------------------------------------- layernorm: wave per row
__global__ void __launch_bounds__(256) ln_kernel(const float* __restrict__ h,
                                                 const float* __restrict__ g,
                                                 const float* __restrict__ b,
                                                 float* __restrict__ hn, int N) {
  int wave = threadIdx.x >> 5;
  int lane = threadIdx.x & 31;
  int row = blockIdx.x * 8 + wave;
  if (row >= N) return;

  float4 v = ((const float4*)(h + (size_t)row * HID))[lane];
  float s = v.x + v.y + v.z + v.w;
#pragma unroll
  for (int off = 16; off > 0; off >>= 1) s += __shfl_xor(s, off, 32);
  float mu = s * (1.0f / HID);

  float dx = v.x - mu, dy = v.y - mu, dz = v.z - mu, dw = v.w - mu;
  float q = dx * dx + dy * dy + dz * dz + dw * dw;
#pragma unroll
  for (int off = 16; off > 0; off >>= 1) q += __shfl_xor(q, off, 32);
  float rstd = rsqrtf(q * (1.0f / HID) + LN_EPS);

  float4 gv = ((const float4*)g)[lane];
  float4 bv = ((const float4*)b)[lane];
  float4 o;
  o.x = dx * rstd * gv.x + bv.x;
  o.y = dy * rstd * gv.y + bv.y;
  o.z = dz * rstd * gv.z + bv.z;
  o.w = dw * rstd * gv.w + bv.w;
  ((float4*)(hn + (size_t)row * HID))[lane] = o;
}

// ------------------------------- edge scatter: wave per edge, f32 atomics
__global__ void __launch_bounds__(256) scatter_kernel(const int* __restrict__ src,
                                                      const int* __restrict__ dst,
                                                      const float* __restrict__ hn,
                                                      float* __restrict__ agg, int E) {
  int gid = blockIdx.x * blockDim.x + threadIdx.x;
  int e = gid >> 5;
  int lane = gid & 31;
  if (e >= E) return;
  int s = src[e];
  int d = dst[e];
  float4 v = ((const float4*)(hn + (size_t)s * HID))[lane];
  float* ap = agg + (size_t)d * HID + lane * 4;
  unsafeAtomicAdd(ap + 0, v.x);
  unsafeAtomicAdd(ap + 1, v.y);
  unsafeAtomicAdd(ap + 2, v.z);
  unsafeAtomicAdd(ap + 3, v.w);
}

// ---------------------------------------------------------------- WMMA GEMM
// h = relu(h + (agg*inv_deg) @ Wl + bl + hn @ Wr), fp32 via V_WMMA_F32_16X16X4_F32.
// One wave per 16x16 output tile; block = 8 waves covering one 16-row stripe
// across all 8 column tiles of HID=128. EXEC stays all-ones (WMMA requirement).
__global__ void __launch_bounds__(256) sage_gemm_kernel(
    const float* __restrict__ agg, const float* __restrict__ hn,
    const float* __restrict__ invdeg, const float* __restrict__ Wl,
    const float* __restrict__ Wr, const float* __restrict__ bl,
    float* __restrict__ h, int N) {
  int wave = threadIdx.x >> 5;
  int lane = threadIdx.x & 31;
  int half = lane >> 4;   // which half of the wave
  int l16 = lane & 15;

  int m0 = blockIdx.x * 16;
  int row = m0 + l16;
  int rowc = row < N ? row : N - 1;       // clamp loads, keep EXEC full
  int col = wave * 16 + l16;

  float invd = invdeg[rowc];
  const float* aggr = agg + (size_t)rowc * HID;
  const float* hnr = hn + (size_t)rowc * HID;

  v8f acc = {};
#pragma unroll 4
  for (int kb = 0; kb < HID; kb += 4) {
    int k0 = kb + 2 * half;
    // A fragment (16x4 f32): lane holds M=l16, K={k0,k0+1}; contiguous load.
    v2f a1 = *(const v2f*)(aggr + k0);
    a1.x *= invd;
    a1.y *= invd;
    // B fragment (4x16 f32): lane holds N=col, K={k0,k0+1} (stride HID apart).
    v2f b1;
    b1.x = Wl[k0 * HID + col];
    b1.y = Wl[(k0 + 1) * HID + col];
    acc = __builtin_amdgcn_wmma_f32_16x16x4_f32(false, a1, false, b1,
                                                (short)0, acc, false, false);
    v2f a2 = *(const v2f*)(hnr + k0);
    v2f b2;
    b2.x = Wr[k0 * HID + col];
    b2.y = Wr[(k0 + 1) * HID + col];
    acc = __builtin_amdgcn_wmma_f32_16x16x4_f32(false, a2, false, b2,
                                                (short)0, acc, false, false);
  }

  float bias = bl[col];
  // C/D layout: VGPR v holds row m0 + v + 8*half, column `col`.
  float* hp = h + (size_t)(m0 + 8 * half) * HID + col;
  if (m0 + 16 <= N) {
    // Whole tile in bounds (always true when N % 16 == 0): branch-free stores.
#pragma unroll
    for (int v = 0; v < 8; ++v) {
      float r = hp[(size_t)v * HID] + acc[v] + bias;
      hp[(size_t)v * HID] = r > 0.0f ? r : 0.0f;
    }
  } else {
#pragma unroll
    for (int v = 0; v < 8; ++v) {
      int m = m0 + v + 8 * half;
      if (m < N) {
        float r = hp[(size_t)v * HID] + acc[v] + bias;
        hp[(size_t)v * HID] = r > 0.0f ? r : 0.0f;
      }
    }
  }
}

// ------------------------------------------------- output projection (N=4)
__global__ void out_proj_kernel(const float* __restrict__ h,
                                const float* __restrict__ W_out,
                                const float* __restrict__ b_out,
                                float* __restrict__ out, int N) {
  int idx = blockIdx.x * blockDim.x + threadIdx.x;
  if (idx >= N * D_OUT) return;
  int row = idx >> 2;
  int c = idx & 3;
  const float* hr = h + (size_t)row * HID;
  float acc = b_out[c];
#pragma unroll 8
  for (int k = 0; k < HID; ++k) acc += hr[k] * W_out[k * D_OUT + c];
  out[idx] = acc;
}

// --------------------------------------------------------------------------
extern "C" void kernel_launch(void* const* d_in, const int* in_sizes, int n_in,
                              void* d_out, int out_size, void* d_ws, size_t ws_size,
                              hipStream_t stream) {
  const float* x = (const float*)d_in[0];
  const int* ei = (const int*)d_in[1];
  const float* W_in = (const float*)d_in[2];
  const float* b_in = (const float*)d_in[3];
  const float* Wl = (const float*)d_in[4];
  const float* bl = (const float*)d_in[5];
  const float* Wr = (const float*)d_in[6];
  const float* ln_g = (const float*)d_in[7];
  const float* ln_b = (const float*)d_in[8];
  const float* W_out = (const float*)d_in[9];
  const float* b_out = (const float*)d_in[10];
  float* out = (float*)d_out;

  int N = in_sizes[0] / D_IN;
  int E = in_sizes[1] / 2;
  const int* src = ei;
  const int* dst = ei + E;

  float* ws = (float*)d_ws;
  size_t nh = (size_t)N * HID;
  float* h = ws;            // [N, HID]
  float* hn = ws + nh;      // [N, HID]
  float* agg = ws + 2 * nh; // [N, HID]
  float* deg = ws + 3 * nh; // [N]  (deg, then inv_deg in place)

  // degrees -> inv_deg
  zero_kernel<<<256, 256, 0, stream>>>(deg, N);
  deg_kernel<<<(E + 255) / 256, 256, 0, stream>>>(dst, deg, E);
  invdeg_kernel<<<(N + 255) / 256, 256, 0, stream>>>(deg, N);

  // h = x @ W_in + b_in
  in_proj_kernel<<<(int)((nh + 255) / 256), 256, 0, stream>>>(x, W_in, b_in, h, N);

  int mtiles = (N + 15) / 16;
  long long scat_threads = (long long)E * 32;
  for (int l = 0; l < 3; ++l) {
    ln_kernel<<<(N + 7) / 8, 256, 0, stream>>>(h, ln_g + l * HID, ln_b + l * HID, hn, N);
    zero_kernel<<<2048, 256, 0, stream>>>(agg, (int)nh);
    scatter_kernel<<<(int)((scat_threads + 255) / 256), 256, 0, stream>>>(src, dst, hn, agg, E);
    sage_gemm_kernel<<<mtiles, 256, 0, stream>>>(agg, hn, deg,
                                                 Wl + (size_t)l * HID * HID,
                                                 Wr + (size_t)l * HID * HID,
                                                 bl + l * HID, h, N);
  }

  out_proj_kernel<<<(N * D_OUT + 255) / 256, 256, 0, stream>>>(h, W_out, b_out, out, N);
}